// NodeEmbedding_84731114815819
// MI455X (gfx1250) — compile-verified
//
#include <hip/hip_runtime.h>

// ---------------------------------------------------------------------------
// NodeEmbedding GCN for MI455X (gfx1250, wave32).
//
// Roofline: edge gather/scatter (2 x 819MB gathers + ~410M f32 atomics)
// bounds this workload at ~100us on 23.3TB/s HBM; GEMMs are ~4.5 GFLOP total,
// negligible. => keep full FP32 precision and use V_WMMA_F32_16X16X4_F32.
// Weights are staged into LDS with the Tensor Data Mover (tensor_load_to_lds,
// TENSORcnt) and repacked into WMMA fragment order so the inner loop is pure
// ds_load_b64 + v_wmma with immediate offsets.
// All row counts are multiples of 16 (50000 = 3125*16) and all feature dims
// multiples of 16, so WMMA tiles are always full and EXEC stays all-ones.
// ---------------------------------------------------------------------------

typedef float v2f __attribute__((ext_vector_type(2)));
typedef float v8f __attribute__((ext_vector_type(8)));
typedef unsigned int u32x4 __attribute__((ext_vector_type(4)));
typedef int i32x4 __attribute__((ext_vector_type(4)));
typedef int i32x8 __attribute__((ext_vector_type(8)));

#define NN 50000
#define EE 1600000
#define DIN 128     // in_dim (= 1 + 127 features)
#define DHID 128
#define DEMB 64
#define DCAT (DIN + DHID + DEMB)   // 320

// ------------------------------- utilities ---------------------------------

__global__ void zero_f32_kernel(float* __restrict__ p, long long n) {
  long long i = (long long)blockIdx.x * blockDim.x + threadIdx.x;
  if (i < n) p[i] = 0.0f;
}

__global__ void degree_kernel(const int* __restrict__ dst,
                              unsigned* __restrict__ deg, int E) {
  int e = blockIdx.x * blockDim.x + threadIdx.x;
  if (e < E) atomicAdd(&deg[dst[e]], 1u);   // integer: deterministic
}

// h1[i] = [deg(i), features(i, 0..126)]
__global__ void build_h1_kernel(const float* __restrict__ feat,
                                const unsigned* __restrict__ deg,
                                float* __restrict__ h1, int n) {
  long long idx = (long long)blockIdx.x * blockDim.x + threadIdx.x;
  if (idx >= (long long)n * DIN) return;
  int i = (int)(idx >> 7);        // /128
  int c = (int)(idx & 127);
  h1[idx] = (c == 0) ? (float)deg[i]
                     : feat[(long long)i * (DIN - 1) + (c - 1)];
}

// S[dst] += H[src]  (mailbox sum), dim fixed to 128.
// One thread per (edge, 16-byte chunk): float4 gather + 4 f32 scatter atomics.
__global__ void agg128_kernel(const float* __restrict__ H,
                              const int* __restrict__ src,
                              const int* __restrict__ dst,
                              float* __restrict__ S, int E) {
  long long t = (long long)blockIdx.x * blockDim.x + threadIdx.x;
  if (t >= (long long)E * 32) return;
  int chunk = (int)(t & 31);
  long long e = t >> 5;
  long long s = src[e];
  long long d = dst[e];
  const float4 v = *((const float4*)(H + s * 128) + chunk);
  float* o = S + d * 128 + (long long)chunk * 4;
  atomicAdd(o + 0, v.x);
  atomicAdd(o + 1, v.y);
  atomicAdd(o + 2, v.z);
  atomicAdd(o + 3, v.w);
}

// In-place: S = deg>0 ? S/deg : H   (DGL zero-in-degree keeps own features)
__global__ void finalize_mean_kernel(float* __restrict__ S,
                                     const float* __restrict__ H,
                                     const unsigned* __restrict__ deg,
                                     int n, int dim) {
  long long idx = (long long)blockIdx.x * blockDim.x + threadIdx.x;
  if (idx >= (long long)n * dim) return;
  int i = (int)(idx / dim);
  unsigned dg = deg[i];
  S[idx] = (dg > 0u) ? S[idx] / (float)dg : H[idx];
}

// ------------------- TDM: stage W[K x F] (f32) into LDS ---------------------
// Builds a D# descriptor per CDNA5 ISA and issues
// tensor_load_to_lds from wave 0 (TDM ignores EXEC; one issue per wave).
// Tile == whole tensor: tensor_dim0=F, tensor_dim1=K, stride0=F, 4B elements.
__device__ __forceinline__ void tdm_stage_w(const float* W, float* sRaw,
                                            int K, int F) {
  if ((threadIdx.x >> 5) == 0) {   // wave 0 only, all 32 lanes active
    unsigned lds_addr = (unsigned)(unsigned long long)sRaw; // addr[31:0] = LDS offset
    unsigned long long ga = (unsigned long long)W;

    u32x4 g0;
    g0.x = 1u;                                          // count=1 (user D#)
    g0.y = lds_addr;                                    // lds_addr[31:0]
    g0.z = (unsigned)(ga & 0xffffffffull);              // global_addr[31:0]
    g0.w = (unsigned)((ga >> 32) & 0x01ffffffull)       // global_addr[56:32]
           | (2u << 30);                                // type=2 ("image")

    i32x8 g1;
    g1[0] = (2 << 16);        // wg_mask=0 | data_size=2 (4 bytes)
    g1[1] = (F << 16);        // tensor_dim0[15:0] in bits 63:48
    g1[2] = (K << 16);        // tensor_dim0[31:16]=0 | tensor_dim1[15:0]
    g1[3] = (F << 16);        // tensor_dim1[31:16]=0 | tile_dim0=F
    g1[4] = K;                // tile_dim1=K | tile_dim2=0
    g1[5] = F;                // tensor_dim0_stride[31:0]
    g1[6] = 0;                // stride0[47:32]=0 | tensor_dim1_stride lo
    g1[7] = 0;

    i32x4 gz = {0, 0, 0, 0};  // groups 2/3 unused (<=2D tensor)
#if defined(__clang_major__) && (__clang_major__ >= 23)
    i32x8 gz8 = {0, 0, 0, 0, 0, 0, 0, 0};
    __builtin_amdgcn_tensor_load_to_lds(g0, g1, gz, gz, gz8, 0);
#else
    __builtin_amdgcn_tensor_load_to_lds(g0, g1, gz, gz, 0);
#endif
    __builtin_amdgcn_s_wait_tensorcnt(0);
  }
  __syncthreads();
}

// Repack row-major sRaw[K*F] into WMMA-fragment order:
// frag[(ct*KS + ks)*32 + j] = { W[4ks+kh][16ct+l], W[4ks+kh+1][16ct+l] }
// with l=j&15, kh=(j>>4)*2 — matches the V_WMMA_F32_16X16X4_F32 B layout, so
// the inner loop reads one ds_load_b64 per k-step at an immediate offset.
__device__ __forceinline__ void repack_w_frags(const float* sRaw, v2f* sFrag,
                                               int K, int F) {
  const int KS = K >> 2;
  const int total = (F >> 4) * KS * 32;
  for (int f = threadIdx.x; f < total; f += blockDim.x) {
    int ct  = f / (KS * 32);
    int rem = f - ct * (KS * 32);
    int ks  = rem >> 5;
    int j   = rem & 31;
    int l   = j & 15;
    int kh  = (j >> 4) << 1;
    int k   = (ks << 2) + kh;
    int c   = (ct << 4) + l;
    v2f b;
    b.x = sRaw[k * F + c];
    b.y = sRaw[(k + 1) * F + c];
    sFrag[f] = b;
  }
  __syncthreads();
}

// ------------------------- WMMA f32 GEMM (+bias,+ReLU) ----------------------
// Y[n,F] = act(X[n,K] @ W[K,F] + bias). One wave32 computes one 16x16 tile;
// block = one 16-row panel, (F/16) waves wide.
//
// V_WMMA_F32_16X16X4_F32 fragment layout (ISA 7.12.2):
//   A 16x4 : lane l(<16) -> row l, {a.x,a.y} = K{0,1}; lane l+16 -> K{2,3}
//   B 4x16 : lane l(<16) -> col l, {b.x,b.y} = K-rows{0,1}; lane l+16 -> {2,3}
//   C 16x16: VGPR r, lane n(<16) -> (M=r, N=n); lane n+16 -> (M=r+8, N=n)

template <int K, int F, bool RELU>
__global__ void gemm_wmma_bias_act(const float* __restrict__ X,
                                   const float* __restrict__ W,
                                   const float* __restrict__ bias,
                                   float* __restrict__ Y) {
  constexpr int KS = K / 4;
  extern __shared__ float lds[];
  float* sRaw  = lds;                       // K*F floats
  v2f*   sFrag = (v2f*)(lds + K * F);       // K*F floats (fragment order)

  tdm_stage_w(W, sRaw, K, F);
  repack_w_frags(sRaw, sFrag, K, F);

  const int lane = threadIdx.x & 31;
  const int wave = threadIdx.x >> 5;        // = col tile
  const int row0 = blockIdx.x * 16;
  const int l    = lane & 15;
  const int kh   = (lane >> 4) << 1;        // 0 or 2

  const v2f* frag = sFrag + (wave * KS) * 32 + lane;     // +32 per k-step
  const float* Arow = X + (long long)(row0 + l) * K + kh;

  v8f acc = {};
#pragma unroll
  for (int ks = 0; ks < KS; ++ks) {
    v2f a = *(const v2f*)(Arow + ks * 4);
    v2f b = frag[ks * 32];
    acc = __builtin_amdgcn_wmma_f32_16x16x4_f32(
        false, a, false, b, (short)0, acc, false, false);
  }

  const float bc = bias[wave * 16 + l];
  const int mbase = (lane >> 4) << 3;       // 0 or 8
#pragma unroll
  for (int r = 0; r < 8; ++r) {
    float v = acc[r] + bc;
    if (RELU) v = fmaxf(v, 0.0f);
    Y[(long long)(row0 + mbase + r) * F + wave * 16 + l] = v;
  }
}

// out[n,64] = [h1 | h2 | h3] @ Wf[320,64] + bf   (no activation)
__global__ void gemm_concat_wmma(const float* __restrict__ h1,
                                 const float* __restrict__ h2,
                                 const float* __restrict__ h3,
                                 const float* __restrict__ Wf,
                                 const float* __restrict__ bf,
                                 float* __restrict__ out) {
  constexpr int KS = DCAT / 4;              // 80 k-steps
  extern __shared__ float lds[];
  float* sRaw  = lds;                       // 320*64 floats
  v2f*   sFrag = (v2f*)(lds + DCAT * DEMB);

  tdm_stage_w(Wf, sRaw, DCAT, DEMB);
  repack_w_frags(sRaw, sFrag, DCAT, DEMB);

  const int lane = threadIdx.x & 31;
  const int wave = threadIdx.x >> 5;        // 4 waves -> F=64
  const int row0 = blockIdx.x * 16;
  const int l    = lane & 15;
  const int kh   = (lane >> 4) << 1;

  const v2f* frag = sFrag + (wave * KS) * 32 + lane;
  const float* A1 = h1 + (long long)(row0 + l) * DIN  + kh;
  const float* A2 = h2 + (long long)(row0 + l) * DHID + kh;
  const float* A3 = h3 + (long long)(row0 + l) * DEMB + kh;

  v8f acc = {};
#pragma unroll
  for (int ks = 0; ks < DIN / 4; ++ks) {    // k in [0,128): h1
    v2f a = *(const v2f*)(A1 + ks * 4);
    acc = __builtin_amdgcn_wmma_f32_16x16x4_f32(
        false, a, false, frag[ks * 32], (short)0, acc, false, false);
  }
#pragma unroll
  for (int ks = 0; ks < DHID / 4; ++ks) {   // k in [128,256): h2
    v2f a = *(const v2f*)(A2 + ks * 4);
    acc = __builtin_amdgcn_wmma_f32_16x16x4_f32(
        false, a, false, frag[(DIN / 4 + ks) * 32], (short)0, acc, false, false);
  }
#pragma unroll
  for (int ks = 0; ks < DEMB / 4; ++ks) {   // k in [256,320): h3
    v2f a = *(const v2f*)(A3 + ks * 4);
    acc = __builtin_amdgcn_wmma_f32_16x16x4_f32(
        false, a, false, frag[((DIN + DHID) / 4 + ks) * 32], (short)0, acc,
        false, false);
  }

  const float bc = bf[wave * 16 + l];
  const int mbase = (lane >> 4) << 3;
#pragma unroll
  for (int r = 0; r < 8; ++r)
    out[(long long)(row0 + mbase + r) * DEMB + wave * 16 + l] = acc[r] + bc;
}

// ------------------------------- launcher ----------------------------------

extern "C" void kernel_launch(void* const* d_in, const int* in_sizes, int n_in,
                              void* d_out, int out_size, void* d_ws, size_t ws_size,
                              hipStream_t stream) {
  const float* feat = (const float*)d_in[0];   // [N, 127]
  const int*   src  = (const int*)d_in[1];     // [E]
  const int*   dst  = (const int*)d_in[2];     // [E]
  const float* W0   = (const float*)d_in[3];   // [128,128]
  const float* b0   = (const float*)d_in[4];
  const float* W1   = (const float*)d_in[5];   // [128,64]
  const float* b1   = (const float*)d_in[6];
  const float* Wf   = (const float*)d_in[7];   // [320,64]
  const float* bfv  = (const float*)d_in[8];
  float* out = (float*)d_out;

  // Workspace (floats): deg(u32)[N] | h1[N*128] | s[N*128] | h2[N*128] | h3[N*64]
  float* ws = (float*)d_ws;
  unsigned* deg = (unsigned*)ws;
  float* h1 = ws + NN;
  float* s  = h1 + (long long)NN * DIN;
  float* h2 = s  + (long long)NN * DHID;
  float* h3 = h2 + (long long)NN * DHID;

  const int TB = 256;
  const long long nd  = (long long)NN * 128;
  const long long eth = (long long)EE * 32;
  const int rowBlocks = NN / 16;               // 3125 exactly

  // deg + scatter buffer zero
  zero_f32_kernel<<<(NN + TB - 1) / TB, TB, 0, stream>>>((float*)deg, NN);
  zero_f32_kernel<<<(int)((nd + TB - 1) / TB), TB, 0, stream>>>(s, nd);

  // in-degrees, h1 = [deg, features]
  degree_kernel<<<(EE + TB - 1) / TB, TB, 0, stream>>>(dst, deg, EE);
  build_h1_kernel<<<(int)((nd + TB - 1) / TB), TB, 0, stream>>>(feat, deg, h1, NN);

  // layer 1: mean-aggregate h1, then ReLU(agg @ W0 + b0) -> h2
  agg128_kernel<<<(int)((eth + TB - 1) / TB), TB, 0, stream>>>(h1, src, dst, s, EE);
  finalize_mean_kernel<<<(int)((nd + TB - 1) / TB), TB, 0, stream>>>(s, h1, deg, NN, DIN);
  gemm_wmma_bias_act<DIN, DHID, true>
      <<<rowBlocks, 32 * (DHID / 16), 2 * DIN * DHID * sizeof(float), stream>>>(
          s, W0, b0, h2);

  // layer 2: mean-aggregate h2, then ReLU(agg @ W1 + b1) -> h3
  zero_f32_kernel<<<(int)((nd + TB - 1) / TB), TB, 0, stream>>>(s, nd);
  agg128_kernel<<<(int)((eth + TB - 1) / TB), TB, 0, stream>>>(h2, src, dst, s, EE);
  finalize_mean_kernel<<<(int)((nd + TB - 1) / TB), TB, 0, stream>>>(s, h2, deg, NN, DHID);
  gemm_wmma_bias_act<DHID, DEMB, true>
      <<<rowBlocks, 32 * (DEMB / 16), 2 * DHID * DEMB * sizeof(float), stream>>>(
          s, W1, b1, h3);

  // out = [h1|h2|h3] @ Wf + bf
  gemm_concat_wmma
      <<<rowBlocks, 32 * (DEMB / 16), 2 * DCAT * DEMB * sizeof(float), stream>>>(
          h1, h2, h3, Wf, bfv, out);
}